// InnerRandomMultiShiftTriple_15736760172612
// MI455X (gfx1250) — compile-verified
//
#include <hip/hip_runtime.h>
#include <stdint.h>

// Problem geometry (fixed by the reference): B=8, C=512, H=W=64
#define HW      4096          // 64*64
#define HALFROW 2048          // HW/2
#define C_IN    512
#define C2      256
#define IN_PLANE   (2097152u)   // 512*4096 = 2^21 elements per batch (input)
#define OUT_PLANE  (3145728u)   // 768*4096 elements per batch (output)

// ---- CDNA5 async global->LDS staging: copy 16 KB into LDS, 256 threads ----
// Generic LDS pointer low 32 bits are the LDS byte address (aperture rule).
__device__ __forceinline__ void stage16k_async(const void* src, void* lds) {
  const uint32_t lds_base = (uint32_t)(uintptr_t)lds;
  const int tid = threadIdx.x;
#pragma unroll
  for (int k = 0; k < 4; ++k) {
    const int      byteoff = k * 4096 + tid * 16;
    const uint64_t ga      = (uint64_t)(uintptr_t)((const char*)src + byteoff);
    const uint32_t la      = lds_base + (uint32_t)byteoff;
    asm volatile("global_load_async_to_lds_b128 %0, %1, off"
                 :: "v"(la), "v"(ga) : "memory");
  }
  asm volatile("s_wait_asynccnt 0x0" ::: "memory");
  __syncthreads();   // publish LDS across all 8 waves in the workgroup
}

// ---------------- JAX threefry2x32 (key = [0, 42]) ----------------
__device__ __forceinline__ uint32_t rotl32(uint32_t x, int r) {
  return (x << r) | (x >> (32 - r));
}

__device__ __forceinline__ void threefry2x32_0_42(uint32_t x0, uint32_t x1,
                                                  uint32_t& y0, uint32_t& y1) {
  const uint32_t ks0 = 0u;
  const uint32_t ks1 = 42u;
  const uint32_t ks2 = 0x1BD11BDAu ^ 0u ^ 42u;
  x0 += ks0; x1 += ks1;
#define TF_R(r) { x0 += x1; x1 = rotl32(x1, (r)); x1 ^= x0; }
  TF_R(13) TF_R(15) TF_R(26) TF_R(6)   x0 += ks1; x1 += ks2 + 1u;
  TF_R(17) TF_R(29) TF_R(16) TF_R(24)  x0 += ks2; x1 += ks0 + 2u;
  TF_R(13) TF_R(15) TF_R(26) TF_R(6)   x0 += ks0; x1 += ks1 + 3u;
  TF_R(17) TF_R(29) TF_R(16) TF_R(24)  x0 += ks1; x1 += ks2 + 4u;
  TF_R(13) TF_R(15) TF_R(26) TF_R(6)   x0 += ks2; x1 += ks0 + 5u;
#undef TF_R
  y0 = x0; y1 = x1;
}

// jax.random.uniform bit conversion: u = bitcast((y>>9)|0x3F800000) - 1.0f, u in [0,1)
__device__ __forceinline__ uint32_t unif_bits(uint32_t y) {
  float f = __uint_as_float((y >> 9) | 0x3F800000u) - 1.0f;
  return __float_as_uint(f);   // non-negative => bit order == float order
}

// ---------------- Kernel 1: per-row argmax of the virtual scores matrix ----
// Block i handles rows (i, i+2048): counter pair (l, l+2^23) yields both.
// Writes cnb[p] = flag[p] ? argmax_j{flag[j]==0} scores[p,j] : -1
__global__ void argmax_nb_kernel(const int* __restrict__ mask,
                                 int* __restrict__ cnb) {
  __shared__ int sflag[HW];                    // 16 KB
  __shared__ unsigned long long red0[256];     // 2 KB
  __shared__ unsigned long long red1[256];     // 2 KB
  const int tid = threadIdx.x;
  const int i   = blockIdx.x;                  // 0..2047

  stage16k_async(mask, sflag);                 // async global->LDS, one wait

  const int fa = sflag[i] > 0;
  const int fb = sflag[i + HALFROW] > 0;

  unsigned long long b0 = 0ull, b1 = 0ull;
  if (fa | fb) {                               // nb unused when both flags are 0
    const uint32_t rowbase = (uint32_t)i << 12;          // i*4096
    for (int j = tid; j < HW; j += 256) {
      if (sflag[j] > 0) continue;              // flagged column => NEG_INF
      uint32_t y0, y1;
      threefry2x32_0_42(rowbase + (uint32_t)j,
                        rowbase + (uint32_t)j + 8388608u, y0, y1);
      const unsigned long long tag = (unsigned long long)(0xFFFFFFFFu - (uint32_t)j);
      const unsigned long long p0 = ((unsigned long long)unif_bits(y0) << 32) | tag;
      const unsigned long long p1 = ((unsigned long long)unif_bits(y1) << 32) | tag;
      if (p0 > b0) b0 = p0;
      if (p1 > b1) b1 = p1;
    }
  }

  red0[tid] = b0; red1[tid] = b1;
  __syncthreads();
  for (int s = 128; s > 0; s >>= 1) {
    if (tid < s) {
      if (red0[tid + s] > red0[tid]) red0[tid] = red0[tid + s];
      if (red1[tid + s] > red1[tid]) red1[tid] = red1[tid + s];
    }
    __syncthreads();
  }

  if (tid == 0) {
    const unsigned long long m0 = red0[0], m1 = red1[0];
    const int nb0 = m0 ? (int)(0xFFFFFFFFu - (uint32_t)m0) : 0;  // all-masked => argmax 0
    const int nb1 = m1 ? (int)(0xFFFFFFFFu - (uint32_t)m1) : 0;
    cnb[i]           = fa ? nb0 : -1;
    cnb[i + HALFROW] = fb ? nb1 : -1;
  }
}

// ---------------- Kernel 2: stream x (all 512 ch) into out channels 0..511 --
__global__ void copy_passthrough_kernel(const float4* __restrict__ x,
                                        float* __restrict__ out) {
  const size_t t = (size_t)blockIdx.x * 256 + threadIdx.x;  // float4 index
  const size_t e = t << 2;                                  // element index
  const size_t b   = e >> 21;                               // / (512*4096)
  const size_t rem = e & (IN_PLANE - 1);
  const float4 v = x[t];
  *(float4*)(out + b * (size_t)OUT_PLANE + rem) = v;
}

// ---------------- Kernel 3: shift = gather via LDS-staged plane -------------
// One block per (b, cs) plane. Stage 16 KB source plane with CDNA5 async
// global->LDS loads, then gather: out = cnb[p] >= 0 ? plane[cnb[p]] : 0.
__global__ void shift_gather_kernel(const float* __restrict__ x,
                                    const int* __restrict__ cnb,
                                    float* __restrict__ out) {
  __shared__ float plane[HW];                  // 16 KB
  const int tid = threadIdx.x;
  const int blk = blockIdx.x;                  // 0..2047
  const int b   = blk >> 8;
  const int cs  = blk & 255;

  const float* src = x   + (((size_t)b << 21) + ((size_t)(C2 + cs) << 12));
  float*       dst = out + ((size_t)b * OUT_PLANE + ((size_t)(C_IN + cs) << 12));

  stage16k_async(src, plane);                  // async global->LDS, one wait

#pragma unroll
  for (int k = 0; k < 4; ++k) {
    const int p = k * 1024 + tid * 4;
    const int4 nb = *(const int4*)(cnb + p);
    float4 r;
    r.x = (nb.x >= 0) ? plane[nb.x] : 0.0f;
    r.y = (nb.y >= 0) ? plane[nb.y] : 0.0f;
    r.z = (nb.z >= 0) ? plane[nb.z] : 0.0f;
    r.w = (nb.w >= 0) ? plane[nb.w] : 0.0f;
    *(float4*)(dst + p) = r;
  }
}

// ---------------------------------------------------------------------------
extern "C" void kernel_launch(void* const* d_in, const int* in_sizes, int n_in,
                              void* d_out, int out_size, void* d_ws, size_t ws_size,
                              hipStream_t stream) {
  (void)in_sizes; (void)n_in; (void)out_size; (void)ws_size;
  const float* x    = (const float*)d_in[0];   // (8, 512, 64, 64) f32
  const int*   mask = (const int*)d_in[1];     // (64, 64) i32
  float*       out  = (float*)d_out;           // (8, 768, 64, 64) f32
  int*         cnb  = (int*)d_ws;              // 4096 ints of scratch

  argmax_nb_kernel       <<<dim3(HALFROW), dim3(256), 0, stream>>>(mask, cnb);
  copy_passthrough_kernel<<<dim3(16384),  dim3(256), 0, stream>>>((const float4*)x, out);
  shift_gather_kernel    <<<dim3(2048),   dim3(256), 0, stream>>>(x, cnb, out);
}